// GCN_4836133175947
// MI455X (gfx1250) — compile-verified
//
#include <hip/hip_runtime.h>
#include <hip/hip_bf16.h>
#include <stdint.h>

typedef float v2f __attribute__((ext_vector_type(2)));
typedef float v8f __attribute__((ext_vector_type(8)));

// ---------------------------------------------------------------------------
// Degree / normalization kernels
// ---------------------------------------------------------------------------
__global__ void deg_init_kernel(float* __restrict__ deg, int n) {
    int i = blockIdx.x * blockDim.x + threadIdx.x;
    if (i < n) deg[i] = 1.0f;  // self-loop contributes 1 to every node's degree
}

__global__ void deg_accum_kernel(const int64_t* __restrict__ dst,
                                 float* __restrict__ deg, long nE) {
    long e = (long)blockIdx.x * blockDim.x + threadIdx.x;
    if (e < nE) atomicAdd(&deg[(size_t)dst[e]], 1.0f);
}

__global__ void deg_rsqrt_kernel(const float* __restrict__ deg,
                                 float* __restrict__ dinv, int n) {
    int i = blockIdx.x * blockDim.x + threadIdx.x;
    if (i < n) {
        float d = deg[i];
        dinv[i] = d > 0.0f ? rsqrtf(d) : 0.0f;
    }
}

// ---------------------------------------------------------------------------
// fp32 WMMA GEMM:  hs[m, :] = (X[m, :] @ W) * dinv[m]   and acc = hs (self-loop)
//   X: [nrows, K] row-major, W: [K, FO] row-major.
//   One wave computes one 16x16 output tile via V_WMMA_F32_16X16X4_F32.
//   Block = 128 threads = 4 waves. FO/16 n-tiles per m-tile.
// ---------------------------------------------------------------------------
template <int K, int FO>
__global__ void gcn_gemm_wmma_kernel(const float* __restrict__ X,
                                     const float* __restrict__ W,
                                     const float* __restrict__ dinv,
                                     float* __restrict__ hs,
                                     float* __restrict__ acc,
                                     int nrows) {
    constexpr int NT = FO / 16;          // n-tiles per row-tile
    constexpr int MT_PER_BLK = 4 / NT;   // 4 waves per block
    const int wave = threadIdx.x >> 5;
    const int lane = threadIdx.x & 31;
    const int mtile = blockIdx.x * MT_PER_BLK + wave / NT;
    const int ntile = wave % NT;
    const int m0 = mtile * 16;
    if (m0 >= nrows) return;             // wave-uniform: EXEC stays all-1s

    const int half = lane >> 4;          // 0: lanes 0-15, 1: lanes 16-31
    const int ln   = lane & 15;
    const int kofs = half * 2;           // A/B K-offset within the 4-wide step
    const int bcol = ntile * 16 + ln;

    const float* xrow = X + (size_t)(m0 + ln) * K;

    v8f c = {};
#pragma unroll
    for (int kk = 0; kk < K; kk += 4) {
        // A 16x4 fp32: VGPR0 = K=kofs, VGPR1 = K=kofs+1 (lanes 0-15: K 0/1,
        // lanes 16-31: K 2/3), row m0+ln striped across lanes.
        v2f a, b;
        a.x = xrow[kk + kofs];
        a.y = xrow[kk + kofs + 1];
        // B 4x16 fp32: row K striped across lanes within a VGPR, mirroring A.
        b.x = W[(size_t)(kk + kofs) * FO + bcol];
        b.y = W[(size_t)(kk + kofs + 1) * FO + bcol];
        c = __builtin_amdgcn_wmma_f32_16x16x4_f32(
                /*neg_a=*/false, a, /*neg_b=*/false, b,
                /*c_mod=*/(short)0, c, /*reuse_a=*/false, /*reuse_b=*/false);
    }

    // C/D layout: VGPR r -> row m0 + r + 8*half, col = bcol.
#pragma unroll
    for (int r = 0; r < 8; ++r) {
        int row = m0 + r + half * 8;
        float v = c[r] * dinv[row];
        size_t idx = (size_t)row * FO + bcol;
        hs[idx]  = v;   // scaled features for gathering
        acc[idx] = v;   // accumulator pre-seeded with the self-loop term
    }
}

// ---------------------------------------------------------------------------
// Edge scatter-add: acc[dst] += hs[src]  (per-edge norm already folded into hs)
// Lane -> (edge, float4 chunk); consecutive lanes cover consecutive chunks.
// ---------------------------------------------------------------------------
template <int F>
__global__ void gcn_scatter_kernel(const float* __restrict__ hs,
                                   float* __restrict__ acc,
                                   const int64_t* __restrict__ src,
                                   const int64_t* __restrict__ dst,
                                   long nE) {
    constexpr int CH = F / 4;            // float4 chunks per edge
    long tid = (long)blockIdx.x * blockDim.x + threadIdx.x;
    if (tid >= nE * CH) return;
    long e = tid / CH;
    int  cp = (int)(tid % CH);
    size_t s = (size_t)src[e];
    size_t d = (size_t)dst[e];
    const float4 v = *(const float4*)(hs + s * F + cp * 4);
    float* base = acc + d * F + cp * 4;
    atomicAdd(base + 0, v.x);
    atomicAdd(base + 1, v.y);
    atomicAdd(base + 2, v.z);
    atomicAdd(base + 3, v.w);
}

// ---------------------------------------------------------------------------
// Epilogues
// ---------------------------------------------------------------------------
template <int F>
__global__ void finalize_relu_kernel(float* __restrict__ acc,   // in-place
                                     const float* __restrict__ dinv,
                                     const float* __restrict__ bias,
                                     long total) {
    long tid = (long)blockIdx.x * blockDim.x + threadIdx.x;
    if (tid >= total) return;
    long i = tid / F;
    int  f = (int)(tid % F);
    acc[tid] = fmaxf(acc[tid] * dinv[i] + bias[f], 0.0f);
}

template <int F>
__global__ void finalize_out_kernel(const float* __restrict__ acc,
                                    const float* __restrict__ dinv,
                                    const float* __restrict__ bias,
                                    float* __restrict__ out,
                                    long total) {
    long tid = (long)blockIdx.x * blockDim.x + threadIdx.x;
    if (tid >= total) return;
    long i = tid / F;
    int  f = (int)(tid % F);
    out[tid] = acc[tid] * dinv[i] + bias[f];
}

// ---------------------------------------------------------------------------
// Launcher
// ---------------------------------------------------------------------------
extern "C" void kernel_launch(void* const* d_in, const int* in_sizes, int n_in,
                              void* d_out, int out_size, void* d_ws, size_t ws_size,
                              hipStream_t stream) {
    const float*   x  = (const float*)d_in[0];
    const int64_t* ei = (const int64_t*)d_in[1];  // [2, E] int64
    const float*   W1 = (const float*)d_in[2];
    const float*   b1 = (const float*)d_in[3];
    const float*   W2 = (const float*)d_in[4];
    const float*   b2 = (const float*)d_in[5];

    const int  FHID = in_sizes[3];                 // 64
    const int  FOUT = in_sizes[5];                 // 16
    const int  FIN  = in_sizes[2] / FHID;          // 128
    const int  N    = in_sizes[0] / FIN;           // 100000
    const long E    = (long)in_sizes[1] / 2;       // 1600000

    const int64_t* srcIdx = ei;
    const int64_t* dstIdx = ei + E;

    // Workspace carve-out (~65 MB):
    float* deg  = (float*)d_ws;                       // N
    float* dinv = deg  + N;                           // N
    float* hs1  = dinv + N;                           // N*FHID
    float* acc1 = hs1  + (size_t)N * FHID;            // N*FHID (later reused as h1)
    float* hs2  = acc1 + (size_t)N * FHID;            // N*FOUT
    float* acc2 = hs2  + (size_t)N * FOUT;            // N*FOUT

    const int T = 256;

    // --- degrees & dinv (shared by both layers) ---
    deg_init_kernel<<<(N + T - 1) / T, T, 0, stream>>>(deg, N);
    deg_accum_kernel<<<(int)((E + T - 1) / T), T, 0, stream>>>(dstIdx, deg, E);
    deg_rsqrt_kernel<<<(N + T - 1) / T, T, 0, stream>>>(deg, dinv, N);

    // --- layer 1: hs1 = (x @ W1) * dinv ; acc1 seeded with self-loop ---
    {
        int mtiles = (N + 15) / 16;                        // 6250, NT=4 -> 1 mtile/blk
        gcn_gemm_wmma_kernel<128, 64><<<mtiles, 128, 0, stream>>>(
            x, W1, dinv, hs1, acc1, N);
        long total = E * (FHID / 4);
        gcn_scatter_kernel<64><<<(int)((total + T - 1) / T), T, 0, stream>>>(
            hs1, acc1, srcIdx, dstIdx, E);
        long tot2 = (long)N * FHID;
        finalize_relu_kernel<64><<<(int)((tot2 + T - 1) / T), T, 0, stream>>>(
            acc1, dinv, b1, tot2);                          // acc1 now holds h1
    }

    // --- layer 2: hs2 = (h1 @ W2) * dinv ; acc2 seeded with self-loop ---
    {
        int mtiles = (N + 15) / 16;                        // NT=1 -> 4 mtiles/blk
        int blocks = (mtiles + 3) / 4;
        gcn_gemm_wmma_kernel<64, 16><<<blocks, 128, 0, stream>>>(
            acc1, W2, dinv, hs2, acc2, N);
        long total = E * (FOUT / 4);
        gcn_scatter_kernel<16><<<(int)((total + T - 1) / T), T, 0, stream>>>(
            hs2, acc2, srcIdx, dstIdx, E);
        long tot2 = (long)N * FOUT;
        finalize_out_kernel<16><<<(int)((tot2 + T - 1) / T), T, 0, stream>>>(
            acc2, dinv, b2, (float*)d_out, tot2);
    }
}